// BackProject_18296560681062
// MI455X (gfx1250) — compile-verified
//
#include <hip/hip_runtime.h>
#include <hip/hip_bf16.h>
#include <stdint.h>

// Problem constants (match reference)
#define B_  2
#define H_  120
#define W_  160
#define C_  32
#define D_  32
#define PIX_PER_IMG (D_ * H_ * W_)      // 614400 sample points per batch image
#define TOTAL_PIX   (B_ * PIX_PER_IMG)  // 1228800
#define PIX_PER_BLOCK 32                // 256 threads, 8 threads (one float4 quad each) per pixel

// Native clang vector (usable with __builtin_nontemporal_store, lowers to b128 ops)
typedef float v4f __attribute__((ext_vector_type(4)));

__global__ __launch_bounds__(256) void backproject_bilerp_kernel(
    const float* __restrict__ fmap,    // [B,H,W,C]
    const float* __restrict__ coords,  // [B,D,H,W,2]
    float* __restrict__ out)           // [B,D,H,W,C]
{
    // Stage this block's 32 coordinate pairs (64 floats = 256B) into LDS via the
    // CDNA5 async global->LDS path. Waves 0..1 issue one b32 each per lane.
    __shared__ float sc[2 * PIX_PER_BLOCK];

    const unsigned tid      = threadIdx.x;
    const unsigned blockPix = blockIdx.x * PIX_PER_BLOCK;

    if (tid < 2u * PIX_PER_BLOCK) {  // waves 0 and 1 fully active (wave32)
        unsigned gofs = (blockPix * 2u + tid) * 4u;            // byte offset into coords
        unsigned lofs = (unsigned)(uintptr_t)(&sc[tid]);       // LDS byte address (low 32 bits)
        asm volatile("global_load_async_to_lds_b32 %0, %1, %2"
                     :: "v"(lofs), "v"(gofs), "s"(coords)
                     : "memory");
        asm volatile("s_wait_asynccnt 0" ::: "memory");
    }
    __syncthreads();

    const int lp = (int)(tid >> 3);        // local pixel 0..31
    const int q  = (int)(tid & 7);         // channel quad 0..7 (4 floats each)
    const int p  = (int)blockPix + lp;     // global sample index in [B*D*H*W)
    const int b  = p / PIX_PER_IMG;        // batch index

    const float x = sc[2 * lp + 0];
    const float y = sc[2 * lp + 1];

    const float x0f = floorf(x);
    const float y0f = floorf(y);
    const float wx  = x - x0f;
    const float wy  = y - y0f;

    int x0 = (int)x0f; x0 = x0 < 0 ? 0 : (x0 > W_ - 1 ? W_ - 1 : x0);
    int x1 = x0 + 1;   x1 = x1 > W_ - 1 ? W_ - 1 : x1;
    int y0 = (int)y0f; y0 = y0 < 0 ? 0 : (y0 > H_ - 1 ? H_ - 1 : y0);
    int y1 = y0 + 1;   y1 = y1 > H_ - 1 ? H_ - 1 : y1;

    const float* img = fmap + (size_t)b * (H_ * W_ * C_);

    // Four coalesced 128-bit gathers (rows are 128B-aligned since C=32 floats).
    const v4f v00 = *((const v4f*)(img + (size_t)(y0 * W_ + x0) * C_) + q);
    const v4f v01 = *((const v4f*)(img + (size_t)(y0 * W_ + x1) * C_) + q);
    const v4f v10 = *((const v4f*)(img + (size_t)(y1 * W_ + x0) * C_) + q);
    const v4f v11 = *((const v4f*)(img + (size_t)(y1 * W_ + x1) * C_) + q);

    const float omx = 1.0f - wx;
    const float omy = 1.0f - wy;

    v4f r;
    r = (v00 * omx + v01 * wx) * omy + (v10 * omx + v11 * wx) * wy;

    // Streaming output (157 MB, never re-read): non-temporal 128-bit store.
    __builtin_nontemporal_store(r, (v4f*)(out + (size_t)p * C_) + q);
}

extern "C" void kernel_launch(void* const* d_in, const int* in_sizes, int n_in,
                              void* d_out, int out_size, void* d_ws, size_t ws_size,
                              hipStream_t stream) {
    const float* fmap   = (const float*)d_in[0];  // [B,H,W,C] float32
    const float* coords = (const float*)d_in[1];  // [B,D,H,W,2] float32
    float* out          = (float*)d_out;          // [B,D,H,W,C] float32

    (void)in_sizes; (void)n_in; (void)out_size; (void)d_ws; (void)ws_size;

    const int blocks = TOTAL_PIX / PIX_PER_BLOCK;  // 38400, exact (no tail)
    backproject_bilerp_kernel<<<blocks, 256, 0, stream>>>(fmap, coords, out);
}